// Slice_73873437491486
// MI455X (gfx1250) — compile-verified
//
#include <hip/hip_runtime.h>
#include <hip/hip_bf16.h>

// Bilateral slice: grid (B=2, GC=12, GD=8, GH=16, GW=16), guide (B,1,1024,1024)
// -> out (B, 12, 1024, 1024), float32 everywhere.

constexpr int GH = 16, GW = 16, GD = 8, GC = 12;
constexpr int IMG_H = 1024, IMG_W = 1024;
constexpr int ROWS_PER_BLOCK = 4;
constexpr int NTHREADS = 256;
constexpr int GRID_ELEMS = GH * GW * GD * GC;           // 24576 floats = 96 KB

#if defined(__gfx1250__) && \
    __has_builtin(__builtin_amdgcn_global_load_async_to_lds_b32) && \
    __has_builtin(__builtin_amdgcn_s_wait_asynccnt)
#define USE_ASYNC_LDS 1
#else
#define USE_ASYNC_LDS 0
#endif

typedef __attribute__((address_space(1))) int global_int_t;
typedef __attribute__((address_space(3))) int lds_int_t;

__global__ __launch_bounds__(NTHREADS) void bilateral_slice_kernel(
    const float* __restrict__ grid_raw,   // (B, GC, GD, GH, GW)
    const float* __restrict__ guide,      // (B, 1, H, W)
    float* __restrict__ out)              // (B, GC, H, W)
{
  // Dynamic LDS, declared as float4 so the base is 16B-aligned for b128 loads.
  extern __shared__ float4 lds4[];
  float* lds = reinterpret_cast<float*>(lds4);

  const int tid   = threadIdx.x;
  const int nrowb = IMG_H / ROWS_PER_BLOCK;
  const int b     = blockIdx.x / nrowb;
  const int i0    = (blockIdx.x % nrowb) * ROWS_PER_BLOCK;

  const float* gsrc = grid_raw + (size_t)b * GRID_ELEMS;

  // ---- Stage grid into LDS with layout transpose:
  //   grid value (c, gk, gi, gj) in gsrc goes to LDS cell (gi, gj, gk), channel c innermost
  // Source side is fully coalesced; LDS side is a per-lane scatter, which the
  // CDNA5 async-to-LDS path supports natively (per-lane LDS dest address).
  // Depth stride in LDS = 12 floats -> the 8 gk values map to 8 distinct banks.
  for (int t = tid; t < GRID_ELEMS; t += NTHREADS) {
    const int gj = t & (GW - 1);
    const int gi = (t >> 4) & (GH - 1);
    const int gk = (t >> 8) & (GD - 1);
    const int c  = t >> 11;
    const int d  = ((gi * GW + gj) * GD + gk) * GC + c;
#if USE_ASYNC_LDS
    __builtin_amdgcn_global_load_async_to_lds_b32(
        (global_int_t*)(gsrc + t),
        (lds_int_t*)(lds + d),
        /*offset=*/0, /*cpol=*/0);
#else
    lds[d] = gsrc[t];
#endif
  }
#if USE_ASYNC_LDS
  __builtin_amdgcn_s_wait_asynccnt(0);   // my wave's async LDS writes landed
#endif
  __syncthreads();                       // cross-wave visibility

  const float* gmap = guide + (size_t)b * IMG_H * IMG_W;
  float* outp       = out   + (size_t)b * GC * IMG_H * IMG_W;

  constexpr float kScaleIJ = (float)GH / (float)IMG_H;  // 1/64
  constexpr float kEps     = 1e-8f;

  for (int p = tid; p < ROWS_PER_BLOCK * IMG_W; p += NTHREADS) {
    const int i = i0 + (p >> 10);        // IMG_W == 1024
    const int j = p & (IMG_W - 1);

    const float g   = gmap[(size_t)i * IMG_W + j];   // coalesced
    const float gif = ((float)i + 0.5f) * kScaleIJ;
    const float gjf = ((float)j + 0.5f) * kScaleIJ;
    const float gkf = g * (float)GD;

    const int gi0 = (int)floorf(gif - 0.5f);
    const int gj0 = (int)floorf(gjf - 0.5f);
    const int gk0 = (int)floorf(gkf - 0.5f);

    // _lerp_w for i/j (distance < 1 by construction, still match reference form)
    float wi[2], wj[2], wk[2];
    wi[0] = fmaxf(1.0f - fabsf(((float)gi0 + 0.5f) - gif), 0.0f);
    wi[1] = fmaxf(1.0f - fabsf(((float)gi0 + 1.5f) - gif), 0.0f);
    wj[0] = fmaxf(1.0f - fabsf(((float)gj0 + 0.5f) - gjf), 0.0f);
    wj[1] = fmaxf(1.0f - fabsf(((float)gj0 + 1.5f) - gjf), 0.0f);
    // _smoothed_lerp_w for k
    {
      const float dx0 = gkf - ((float)gk0 + 0.5f);
      const float dx1 = gkf - ((float)gk0 + 1.5f);
      wk[0] = fmaxf(1.0f - sqrtf(dx0 * dx0 + kEps), 0.0f);
      wk[1] = fmaxf(1.0f - sqrtf(dx1 * dx1 + kEps), 0.0f);
    }

    int ic[2], jc[2], kc[2];
    ic[0] = min(max(gi0, 0), GH - 1);  ic[1] = min(max(gi0 + 1, 0), GH - 1);
    jc[0] = min(max(gj0, 0), GW - 1);  jc[1] = min(max(gj0 + 1, 0), GW - 1);
    kc[0] = min(max(gk0, 0), GD - 1);  kc[1] = min(max(gk0 + 1, 0), GD - 1);

    float acc[GC];
#pragma unroll
    for (int c = 0; c < GC; ++c) acc[c] = 0.0f;

#pragma unroll
    for (int a = 0; a < 2; ++a) {
#pragma unroll
      for (int b2 = 0; b2 < 2; ++b2) {
        const float wij    = wi[a] * wj[b2];
        const int   baseij = (ic[a] * GW + jc[b2]) * GD;
#pragma unroll
        for (int k2 = 0; k2 < 2; ++k2) {
          const float w = wij * wk[k2];
          // Cell = 12 floats = 48 B, 16B-aligned -> three ds_load_b128.
          const float4* cell =
              reinterpret_cast<const float4*>(lds + (baseij + kc[k2]) * GC);
          const float4 v0 = cell[0];
          const float4 v1 = cell[1];
          const float4 v2 = cell[2];
          float vv[GC];
          vv[0] = v0.x; vv[1] = v0.y; vv[2]  = v0.z; vv[3]  = v0.w;
          vv[4] = v1.x; vv[5] = v1.y; vv[6]  = v1.z; vv[7]  = v1.w;
          vv[8] = v2.x; vv[9] = v2.y; vv[10] = v2.z; vv[11] = v2.w;
#pragma unroll
          for (int c = 0; c < GC; ++c) acc[c] = fmaf(w, vv[c], acc[c]);
        }
      }
    }

    const size_t o = (size_t)i * IMG_W + j;
#pragma unroll
    for (int c = 0; c < GC; ++c)
      outp[(size_t)c * IMG_H * IMG_W + o] = acc[c];   // coalesced per channel
  }
}

extern "C" void kernel_launch(void* const* d_in, const int* in_sizes, int n_in,
                              void* d_out, int out_size, void* d_ws, size_t ws_size,
                              hipStream_t stream) {
  (void)in_sizes; (void)n_in; (void)out_size; (void)d_ws; (void)ws_size;
  const float* grid  = (const float*)d_in[0];   // (2,12,8,16,16)
  const float* guide = (const float*)d_in[1];   // (2,1,1024,1024)
  float*       out   = (float*)d_out;           // (2,12,1024,1024)

  const int B = 2;
  const size_t shmem = (size_t)GRID_ELEMS * sizeof(float);   // 96 KB dynamic LDS
  (void)hipFuncSetAttribute((const void*)bilateral_slice_kernel,
                            hipFuncAttributeMaxDynamicSharedMemorySize, (int)shmem);

  dim3 blocks(B * (IMG_H / ROWS_PER_BLOCK));   // 512 blocks
  bilateral_slice_kernel<<<blocks, NTHREADS, shmem, stream>>>(grid, guide, out);
}